// TripletLoss_14233521619194
// MI455X (gfx1250) — compile-verified
//
#include <hip/hip_runtime.h>

typedef __attribute__((ext_vector_type(2))) float v2f;
typedef __attribute__((ext_vector_type(8))) float v8f;

#define BATCH 512
#define DIM   128
#define MARGIN 0.5f
#define CLAMP_MIN 1e-8f

// ---------------------------------------------------------------------------
// Kernel A: squared row norms. One 128-thread block per row.
// ---------------------------------------------------------------------------
__global__ __launch_bounds__(128) void tl_sqnorm_kernel(const float* __restrict__ x,
                                                        float* __restrict__ sqn) {
    __shared__ float s[128];
    const int row = blockIdx.x;
    const int t = threadIdx.x;
    float v = x[row * DIM + t];
    s[t] = v * v;
    __syncthreads();
    for (int off = 64; off > 0; off >>= 1) {
        if (t < off) s[t] += s[t + off];
        __syncthreads();
    }
    if (t == 0) sqn[row] = s[0];
}

// ---------------------------------------------------------------------------
// Kernel B: Gram tile via V_WMMA_F32_16X16X4_F32, fused into distance matrix.
// One wave (32 threads) per 16x16 output tile; 32 WMMA steps over K=128.
//
// A (16x4 f32) per-lane layout: m = lane&15 selects the row, kh = lane>>4
// selects K-halves: VGPR0 = K(2*kh), VGPR1 = K(2*kh+1). For G = X*X^T the
// B operand (4x16 = X^T tile) uses the same per-lane pattern on the column
// tile's rows.
// C/D (16x16 f32, 8 VGPRs): VGPR r holds row (r + 8*kh), lane m is the col.
// ---------------------------------------------------------------------------
__global__ __launch_bounds__(32) void tl_gram_dist_kernel(const float* __restrict__ x,
                                                          const float* __restrict__ sqn,
                                                          float* __restrict__ dist) {
    const int ti = blockIdx.y;          // row tile
    const int tj = blockIdx.x;          // col tile
    const int lane = threadIdx.x;       // 0..31, wave32
    const int m  = lane & 15;
    const int kh = lane >> 4;

    const float* __restrict__ arow = x + (ti * 16 + m) * DIM;
    const float* __restrict__ brow = x + (tj * 16 + m) * DIM;

    v8f c = {};
    #pragma unroll 4
    for (int kk = 0; kk < DIM; kk += 4) {
        v2f a, b;
        a.x = arow[kk + 2 * kh + 0];
        a.y = arow[kk + 2 * kh + 1];
        b.x = brow[kk + 2 * kh + 0];
        b.y = brow[kk + 2 * kh + 1];
        // (neg_a, A, neg_b, B, c_mod, C, reuse_a, reuse_b)
        c = __builtin_amdgcn_wmma_f32_16x16x4_f32(false, a, false, b,
                                                  (short)0, c, false, false);
    }

    const int i0 = ti * 16 + kh * 8;
    const int j  = tj * 16 + m;
    const float sj = sqn[j];
    #pragma unroll
    for (int r = 0; r < 8; ++r) {
        const int i = i0 + r;
        float sq = sqn[i] + sj - 2.0f * c[r];
        sq = fmaxf(sq, 0.0f);
        dist[i * BATCH + j] = (sq > 0.0f) ? sqrtf(sq) : 0.0f;
    }
}

// ---------------------------------------------------------------------------
// Kernel C: per-anchor triplet accumulation. One 256-thread block per anchor.
// dist row + labels staged in LDS; each thread owns negatives {t, t+256};
// deterministic LDS tree reduction -> partial[a].
// ---------------------------------------------------------------------------
__global__ __launch_bounds__(256) void tl_triplet_kernel(const float* __restrict__ dist,
                                                         const int* __restrict__ y,
                                                         float* __restrict__ partial) {
    __shared__ float drow[BATCH];
    __shared__ int   yl[BATCH];
    __shared__ float red[256];

    const int a = blockIdx.x;
    const int t = threadIdx.x;

    drow[t]       = dist[a * BATCH + t];
    drow[t + 256] = dist[a * BATCH + t + 256];
    yl[t]       = y[t];
    yl[t + 256] = y[t + 256];
    __syncthreads();

    const int   ya  = yl[a];
    const float dn0 = drow[t];
    const float dn1 = drow[t + 256];
    const bool  n0  = (yl[t]       != ya);
    const bool  n1  = (yl[t + 256] != ya);

    float acc = 0.0f;
    for (int p = 0; p < BATCH; ++p) {
        if (yl[p] == ya && p != a) {
            const float dpm = drow[p] + MARGIN;
            if (n0) acc += fmaxf(dpm - dn0, CLAMP_MIN);
            if (n1) acc += fmaxf(dpm - dn1, CLAMP_MIN);
        }
    }

    red[t] = acc;
    __syncthreads();
    for (int off = 128; off > 0; off >>= 1) {
        if (t < off) red[t] += red[t + off];
        __syncthreads();
    }
    if (t == 0) partial[a] = red[0];
}

// ---------------------------------------------------------------------------
// Kernel D: deterministic final reduction of 512 partials -> scalar loss.
// ---------------------------------------------------------------------------
__global__ __launch_bounds__(256) void tl_finalize_kernel(const float* __restrict__ partial,
                                                          float* __restrict__ out) {
    __shared__ float red[256];
    const int t = threadIdx.x;
    red[t] = partial[t] + partial[t + 256];
    __syncthreads();
    for (int off = 128; off > 0; off >>= 1) {
        if (t < off) red[t] += red[t + off];
        __syncthreads();
    }
    if (t == 0) out[0] = red[0];
}

extern "C" void kernel_launch(void* const* d_in, const int* in_sizes, int n_in,
                              void* d_out, int out_size, void* d_ws, size_t ws_size,
                              hipStream_t stream) {
    const float* x = (const float*)d_in[0];   // [512,128] f32
    const int*   y = (const int*)d_in[1];     // [512] i32
    float* out = (float*)d_out;               // scalar

    float* dist    = (float*)d_ws;            // 512*512 f32 = 1 MiB
    float* sqn     = dist + BATCH * BATCH;    // 512 f32
    float* partial = sqn + BATCH;             // 512 f32

    tl_sqnorm_kernel<<<BATCH, 128, 0, stream>>>(x, sqn);

    dim3 grid(BATCH / 16, BATCH / 16);        // 32x32 tiles, one wave each
    tl_gram_dist_kernel<<<grid, 32, 0, stream>>>(x, sqn, dist);

    tl_triplet_kernel<<<BATCH, 256, 0, stream>>>(dist, y, partial);

    tl_finalize_kernel<<<1, 256, 0, stream>>>(partial, out);
}